// PrototypeComputation_14972255994268
// MI455X (gfx1250) — compile-verified
//
#include <hip/hip_runtime.h>
#include <math.h>

// Prototype computation (semantic-weighted, per-class softmax) for MI455X.
//
// Shapes (from reference): F = support_features (1000*128, 512) f32,
// labels = repeat(arange(1000), 128), sem = class_semantics (1000, 512) f32,
// out = prototypes (1000, 512) f32.
//
// Strategy: one 1024-thread workgroup (32 wave32) per class.
//  1) Stage the class anchor s_c (2 KB) in LDS, compute ||s_c||.
//  2) Stage the class tile F_c (128x512 f32 = 256 KB) into LDS (CDNA5 WGP has
//     320 KB LDS) while computing per-row dot/||f_r|| -> cosine sims.
//     This is the ONLY HBM pass over F (262 MB total @ 23.3 TB/s ~= 11 us floor).
//  3) Numerically-stable 128-way softmax -> weights w in LDS.
//  4) proto = w^T (1x128) * F_c (128x512) via V_WMMA_F32_16X16X4_F32:
//     each wave owns one 16-column tile of D=512, accumulates 32 K-chunks.
//     A = w broadcast over all 16 M rows, B = F chunk from LDS. Row M=0 of the
//     C accumulator (VGPR0, lanes 0..15) is the prototype slice.

typedef __attribute__((ext_vector_type(2))) float v2f;
typedef __attribute__((ext_vector_type(8))) float v8f;

#define DM  512      // D_MODEL
#define KS  128      // K_SHOT (rows per class)
#define EPSN 1e-8f

__global__ __launch_bounds__(1024) void proto_wmma_kernel(
    const float* __restrict__ sf,      // (n_way*KS, DM)
    const int*   __restrict__ labels,  // (n_way*KS)
    const float* __restrict__ sem,     // (n_way, DM)
    float*       __restrict__ out)     // (n_way, DM)
{
    extern __shared__ float smem[];
    float* Flds = smem;               // KS*DM floats (256 KB): class tile
    float* slds = Flds + KS * DM;     // DM floats: class anchor
    float* sims = slds + DM;          // KS floats: cosine sims
    float* wlds = sims + KS;          // KS floats: softmax weights
    float* red  = wlds + KS;          // 64 floats: reduction scratch

    const int tid  = threadIdx.x;
    const int lane = tid & 31;
    const int wid  = tid >> 5;        // wave id 0..31
    const int blk  = blockIdx.x;
    const int cls  = labels[(size_t)blk * KS];   // class id of this block

    // ---- Phase 1: stage anchor, compute ||s_c||^2 ----
    if (tid == 0) red[0] = 0.0f;
    __syncthreads();
    if (tid < DM / 4) {
        const float4 sv4 = ((const float4*)(sem + (size_t)cls * DM))[tid];
        ((float4*)slds)[tid] = sv4;
        atomicAdd(&red[0], sv4.x*sv4.x + sv4.y*sv4.y + sv4.z*sv4.z + sv4.w*sv4.w);
    }
    __syncthreads();
    const float sn = fmaxf(sqrtf(red[0]), EPSN);

    // ---- Phase 2: stage F_c rows + cosine sims (4 rows per wave) ----
    const float4* s4 = (const float4*)slds;
    for (int i = 0; i < 4; ++i) {
        const int r = wid * 4 + i;
        const float4* frow = (const float4*)(sf + ((size_t)blk * KS + r) * DM);
        float4* lrow = (float4*)(Flds + r * DM);
        float dot = 0.0f, nn = 0.0f;
        #pragma unroll
        for (int j = 0; j < 4; ++j) {              // 512 floats / (32 lanes * 4)
            const int c4 = j * 32 + lane;          // coalesced 512B per wave
            const float4 f = frow[c4];
            lrow[c4] = f;
            const float4 s = s4[c4];
            dot += f.x*s.x + f.y*s.y + f.z*s.z + f.w*s.w;
            nn  += f.x*f.x + f.y*f.y + f.z*f.z + f.w*f.w;
        }
        #pragma unroll
        for (int off = 16; off >= 1; off >>= 1) {  // wave32 tree reduce
            dot += __shfl_xor(dot, off, 32);
            nn  += __shfl_xor(nn,  off, 32);
        }
        if (lane == 0)
            sims[r] = dot / (fmaxf(sqrtf(nn), EPSN) * sn);
    }
    __syncthreads();

    // ---- Phase 3: stable softmax over the KS sims (waves 0..3 carry data) ----
    const float sv = (tid < KS) ? sims[tid] : -INFINITY;
    float mv = sv;
    #pragma unroll
    for (int off = 16; off >= 1; off >>= 1)
        mv = fmaxf(mv, __shfl_xor(mv, off, 32));
    if (lane == 0 && wid < 4) red[8 + wid] = mv;
    __syncthreads();
    const float m = fmaxf(fmaxf(red[8], red[9]), fmaxf(red[10], red[11]));
    const float ev = (tid < KS) ? expf(sv - m) : 0.0f;
    float zv = ev;
    #pragma unroll
    for (int off = 16; off >= 1; off >>= 1)
        zv += __shfl_xor(zv, off, 32);
    if (lane == 0 && wid < 4) red[16 + wid] = zv;
    __syncthreads();
    const float z = red[16] + red[17] + red[18] + red[19];
    if (tid < KS) wlds[tid] = ev / z;
    __syncthreads();

    // ---- Phase 4: proto = w^T * F_c via V_WMMA_F32_16X16X4_F32 ----
    // Wave `wid` owns output columns [wid*16, wid*16+16).
    // A (16x4 f32): lanes 0-15 hold K={0,1} in v0/v1; lanes 16-31 K={2,3}.
    //   Broadcast w over all 16 M rows -> every C row equals the prototype.
    // B (4x16 f32): lanes 0-15 = K={0,1}, lanes 16-31 = K={2,3}; N = lane&15.
    const int d0  = wid * 16;
    const int hi  = lane >> 4;          // 0 -> local K {0,1}, 1 -> {2,3}
    const int col = d0 + (lane & 15);
    v8f acc = {};
    for (int kc = 0; kc < KS / 4; ++kc) {
        const int k0 = kc * 4 + hi * 2;
        v2f a, b;
        a.x = wlds[k0];
        a.y = wlds[k0 + 1];
        b.x = Flds[(size_t)k0 * DM + col];
        b.y = Flds[(size_t)(k0 + 1) * DM + col];
        acc = __builtin_amdgcn_wmma_f32_16x16x4_f32(
            /*neg_a=*/false, a, /*neg_b=*/false, b,
            /*c_mod=*/(short)0, acc, /*reuse_a=*/false, /*reuse_b=*/false);
    }
    // C layout: VGPR0 = row M=0 on lanes 0..15 (N = lane).
    if (lane < 16)
        out[(size_t)cls * DM + d0 + lane] = acc[0];
}

extern "C" void kernel_launch(void* const* d_in, const int* in_sizes, int n_in,
                              void* d_out, int out_size, void* d_ws, size_t ws_size,
                              hipStream_t stream) {
    const float* sf     = (const float*)d_in[0];   // support_features
    const int*   labels = (const int*)d_in[1];     // support_labels
    // d_in[2] = n_way scalar (unused; derived from shapes)
    const float* sem    = (const float*)d_in[3];   // class_semantics
    float* out          = (float*)d_out;

    const int n_way = in_sizes[3] / DM;            // class_semantics is (n_way, DM)

    const size_t lds_bytes =
        (size_t)(KS * DM + DM + KS + KS + 64) * sizeof(float);  // ~259 KB < 320 KB/WGP

    hipFuncSetAttribute((const void*)proto_wmma_kernel,
                        hipFuncAttributeMaxDynamicSharedMemorySize,
                        (int)lds_bytes);

    proto_wmma_kernel<<<n_way, 1024, lds_bytes, stream>>>(sf, labels, sem, out);
}